// TensorProduct_16999480557864
// MI455X (gfx1250) — compile-verified
//
#include <hip/hip_runtime.h>

typedef float v2f __attribute__((ext_vector_type(2)));
typedef float v8f __attribute__((ext_vector_type(8)));

#define ZZ 128
#define DIMF 192
#define WROW 459776
// W_OFFS: w1=0, w2=262144, w3=327680, w4=393216, w5=458752

__global__ __launch_bounds__(512)
void tp_kernel(const float* __restrict__ x1, const float* __restrict__ x2,
               const float* __restrict__ w,  const float* __restrict__ cg,
               float* __restrict__ out)
{
    __shared__ float x1l[DIMF];
    __shared__ float x2l[DIMF];
    __shared__ float P1l[4096];                 // x1s[u]*x2s[v]
    __shared__ float P4l[1024];                 // 0.5*dot(x1v[u],x2v[v])
    __shared__ __align__(16) float Q2l[8192];   // 0.5*x1s[u]*x2v[v][j]   [p=u*32+v][j]
    __shared__ __align__(16) float Q3l[8192];   // 0.5*x1v[u][i]*x2s[v]   [p=u*64+v][i]
    __shared__ float Rl[512];                   // R[v][i][k] = sum_j cg[i,j,k]*x2v[v][j]
    __shared__ float acc[DIMF];

    const int z    = blockIdx.x;
    const int t    = threadIdx.x;
    const int lane = t & 31;
    const int wave = t >> 5;                    // 16 waves

    const float* xz1 = x1 + z * DIMF;
    const float* xz2 = x2 + z * DIMF;
    const float* wz  = w + (size_t)z * WROW;
    const float* w1z = wz;
    const float* w2z = wz + 262144;
    const float* w3z = wz + 327680;
    const float* w4z = wz + 393216;
    const float* w5z = wz + 458752;

    if (t < DIMF) { x1l[t] = xz1[t]; x2l[t] = xz2[t]; }
    __syncthreads();

    // ---- precompute small factor tables in LDS (uniform loops, no divergence) ----
    for (int i = t; i < 4096; i += 512)
        P1l[i] = x1l[i >> 6] * x2l[i & 63];
    for (int i = t; i < 1024; i += 512) {
        int u = i >> 5, v = i & 31;
        float s = 0.f;
        #pragma unroll
        for (int d = 0; d < 4; ++d) s += x1l[64 + u*4 + d] * x2l[64 + v*4 + d];
        P4l[i] = 0.5f * s;
    }
    for (int i = t; i < 8192; i += 512) {
        int p = i >> 2, j = i & 3, u = p >> 5, v = p & 31;
        Q2l[i] = 0.5f * x1l[u] * x2l[64 + v*4 + j];
    }
    for (int i = t; i < 8192; i += 512) {
        int p = i >> 2, ii = i & 3, u = p >> 6, v = p & 63;
        Q3l[i] = 0.5f * x1l[64 + u*4 + ii] * x2l[v];
    }
    for (int i = t; i < 512; i += 512) {
        int v = i >> 4, r = i & 15, ii = r >> 2, k = r & 3;
        float s = 0.f;
        #pragma unroll
        for (int j = 0; j < 4; ++j) s += cg[ii*16 + j*4 + k] * x2l[64 + v*4 + j];
        Rl[i] = s;
    }
    if (t < DIMF) acc[t] = 0.f;
    __syncthreads();

    // ---- Phase 1+2: out_s via V_WMMA_F32_16X16X4_F32 (EXEC all ones here) ----
    // A (16x4 f32): lanes0-15 hold K={0,1}, lanes16-31 hold K={2,3}; rows replicated with p-chunk.
    // B (4x16 f32): VGPR0 = rows K0|K2 striped over lane halves, VGPR1 = rows K1|K3.
    // D rows all equal => row 0 (lanes 0-15 of first acc VGPR) is the out_s partial.
    const int half2 = (lane >> 4) << 1;   // 0 or 2
    const int col   = lane & 15;
    v8f c0 = {0,0,0,0,0,0,0,0}, c1 = {0,0,0,0,0,0,0,0};
    v8f c2 = {0,0,0,0,0,0,0,0}, c3 = {0,0,0,0,0,0,0,0};

    // w1 term: 4096 uv pairs -> 1024 chunks of K=4, rows of 64 w's
    for (int k = wave; k < 1024; k += 16) {
        const int cc = k * 4;
        v2f a; a.x = P1l[cc + half2]; a.y = P1l[cc + half2 + 1];
        const float* bp = w1z + (cc + half2) * 64 + col;
        __builtin_prefetch(bp + 4096, 0, 0);    // 16KB ahead (next iter of this wave)
        v2f b0, b1, b2, b3;
        b0.x = bp[ 0]; b0.y = bp[ 64];
        b1.x = bp[16]; b1.y = bp[ 80];
        b2.x = bp[32]; b2.y = bp[ 96];
        b3.x = bp[48]; b3.y = bp[112];
        c0 = __builtin_amdgcn_wmma_f32_16x16x4_f32(false, a, false, b0, (short)0, c0, false, false);
        c1 = __builtin_amdgcn_wmma_f32_16x16x4_f32(false, a, false, b1, (short)0, c1, false, false);
        c2 = __builtin_amdgcn_wmma_f32_16x16x4_f32(false, a, false, b2, (short)0, c2, false, false);
        c3 = __builtin_amdgcn_wmma_f32_16x16x4_f32(false, a, false, b3, (short)0, c3, false, false);
    }
    // w4 term: 1024 uv pairs -> 256 chunks, same row shape (64 w's), same accumulators
    for (int k = wave; k < 256; k += 16) {
        const int cc = k * 4;
        v2f a; a.x = P4l[cc + half2]; a.y = P4l[cc + half2 + 1];
        const float* bp = w4z + (cc + half2) * 64 + col;
        v2f b0, b1, b2, b3;
        b0.x = bp[ 0]; b0.y = bp[ 64];
        b1.x = bp[16]; b1.y = bp[ 80];
        b2.x = bp[32]; b2.y = bp[ 96];
        b3.x = bp[48]; b3.y = bp[112];
        c0 = __builtin_amdgcn_wmma_f32_16x16x4_f32(false, a, false, b0, (short)0, c0, false, false);
        c1 = __builtin_amdgcn_wmma_f32_16x16x4_f32(false, a, false, b1, (short)0, c1, false, false);
        c2 = __builtin_amdgcn_wmma_f32_16x16x4_f32(false, a, false, b2, (short)0, c2, false, false);
        c3 = __builtin_amdgcn_wmma_f32_16x16x4_f32(false, a, false, b3, (short)0, c3, false, false);
    }
    if (lane < 16) {  // D layout: VGPR0, lanes0-15 = (M=0, N=lane)
        atomicAdd(&acc[     lane], c0[0]);
        atomicAdd(&acc[16 + lane], c1[0]);
        atomicAdd(&acc[32 + lane], c2[0]);
        atomicAdd(&acc[48 + lane], c3[0]);
    }

    // ---- Phase 3: w2 term (VALU streaming, lane = w, wave-uniform LDS broadcast) ----
    {
        const int wi = lane;
        float s0 = 0.f, s1 = 0.f, s2 = 0.f, s3 = 0.f;
        for (int k = 0; k < 128; ++k) {
            const int uv = wave + 16 * k;                    // p = u*32+v, 0..2047
            const float wgt = w2z[uv * 32 + wi];             // 128B coalesced per wave
            const float4 q = *(const float4*)&Q2l[uv * 4];   // uniform -> LDS broadcast
            s0 += wgt * q.x; s1 += wgt * q.y; s2 += wgt * q.z; s3 += wgt * q.w;
        }
        atomicAdd(&acc[64 + wi*4 + 0], s0);
        atomicAdd(&acc[64 + wi*4 + 1], s1);
        atomicAdd(&acc[64 + wi*4 + 2], s2);
        atomicAdd(&acc[64 + wi*4 + 3], s3);
    }

    // ---- Phase 4: w3 term ----
    {
        const int wi = lane;
        float s0 = 0.f, s1 = 0.f, s2 = 0.f, s3 = 0.f;
        for (int k = 0; k < 128; ++k) {
            const int uv = wave + 16 * k;                    // p = u*64+v, 0..2047
            const float wgt = w3z[uv * 32 + wi];
            const float4 q = *(const float4*)&Q3l[uv * 4];
            s0 += wgt * q.x; s1 += wgt * q.y; s2 += wgt * q.z; s3 += wgt * q.w;
        }
        atomicAdd(&acc[64 + wi*4 + 0], s0);
        atomicAdd(&acc[64 + wi*4 + 1], s1);
        atomicAdd(&acc[64 + wi*4 + 2], s2);
        atomicAdd(&acc[64 + wi*4 + 3], s3);
    }

    // ---- Phase 5: w5 * cg term (tiny) ----
    if (t < 128) {
        const int u = t >> 2, k = t & 3;
        float s = 0.f;
        for (int v = 0; v < 32; ++v) {
            float tmp = 0.f;
            #pragma unroll
            for (int i = 0; i < 4; ++i) tmp += x1l[64 + u*4 + i] * Rl[v*16 + i*4 + k];
            s += w5z[u*32 + v] * tmp;
        }
        atomicAdd(&acc[64 + u*4 + k], s);
    }

    __syncthreads();
    if (t < DIMF) out[z * DIMF + t] = acc[t];
}

extern "C" void kernel_launch(void* const* d_in, const int* in_sizes, int n_in,
                              void* d_out, int out_size, void* d_ws, size_t ws_size,
                              hipStream_t stream) {
    const float* x1 = (const float*)d_in[0];
    const float* x2 = (const float*)d_in[1];
    const float* w  = (const float*)d_in[2];
    const float* cg = (const float*)d_in[3];
    float* out = (float*)d_out;
    (void)in_sizes; (void)n_in; (void)out_size; (void)d_ws; (void)ws_size;
    tp_kernel<<<dim3(ZZ), dim3(512), 0, stream>>>(x1, x2, w, cg, out);
}